// Classifier_14714557956356
// MI455X (gfx1250) — compile-verified
//
#include <hip/hip_runtime.h>

#define NN 100000
#define EE 3200000
#define HH 128
#define GG 128
#define CC 10

typedef __attribute__((ext_vector_type(2))) float v2f;
typedef __attribute__((ext_vector_type(8))) float v8f;

__device__ __forceinline__ void atomAddF(float* p, float v) {
    // relaxed, device-scope fp32 atomic -> global_atomic_add_f32 on gfx1250
    __hip_atomic_fetch_add(p, v, __ATOMIC_RELAXED, __HIP_MEMORY_SCOPE_AGENT);
}

// ---------- zero scratch regions (must be re-zeroed every launch) ----------
__global__ void zero_kernel(float* __restrict__ p, long n) {
    long i = (long)blockIdx.x * blockDim.x + threadIdx.x;
    long stride = (long)gridDim.x * blockDim.x;
    for (; i < n; i += stride) p[i] = 0.0f;
}

// ---------- degree computation ----------
__global__ void degree_kernel(const int* __restrict__ src, const int* __restrict__ dst,
                              float* __restrict__ in_deg, float* __restrict__ out_deg) {
    int e = blockIdx.x * blockDim.x + threadIdx.x;
    if (e < EE) {
        atomAddF(&in_deg[dst[e]], 1.0f);
        atomAddF(&out_deg[src[e]], 1.0f);
    }
}

// ---------- norms + layer-1 "hs" scalar (h0 = in_deg) ----------
__global__ void norm_kernel(const float* __restrict__ in_deg, const float* __restrict__ out_deg,
                            float* __restrict__ norm_src, float* __restrict__ norm_dst,
                            float* __restrict__ hs0) {
    int n = blockIdx.x * blockDim.x + threadIdx.x;
    if (n < NN) {
        float od = out_deg[n];  od = od < 1.0f ? 1.0f : od;
        float id = in_deg[n];
        float idc = id < 1.0f ? 1.0f : id;
        float ns = rsqrtf(od);
        float nd = rsqrtf(idc);
        norm_src[n] = ns;
        norm_dst[n] = nd;
        hs0[n] = id * ns;   // h0 * norm_src (h0 = unclipped in_deg)
    }
}

// ---------- layer-1 scalar edge aggregation ----------
__global__ void edge1_kernel(const int* __restrict__ src, const int* __restrict__ dst,
                             const float* __restrict__ hs0, float* __restrict__ agg1) {
    int e = blockIdx.x * blockDim.x + threadIdx.x;
    if (e < EE) atomAddF(&agg1[dst[e]], hs0[src[e]]);
}

// ---------- layer-1 rank-1 expansion + relu + pre-scale by norm_src for layer 2 ----------
__global__ void h1_kernel(const float* __restrict__ agg1,
                          const float* __restrict__ norm_dst, const float* __restrict__ norm_src,
                          const float* __restrict__ W1, const float* __restrict__ b1,
                          float* __restrict__ hs2) {
    long i = (long)blockIdx.x * blockDim.x + threadIdx.x;   // exactly N*H threads
    int n = (int)(i >> 7);
    int j = (int)(i & 127);
    float t = agg1[n] * norm_dst[n];
    float v = fmaf(t, W1[j], b1[j]);
    v = v > 0.0f ? v : 0.0f;
    hs2[i] = v * norm_src[n];           // (relu(...)) * norm_src, ready for edge gather
}

// ---------- layer-2 vector edge aggregation: one wave32 per edge ----------
__global__ void edge2_kernel(const int* __restrict__ src, const int* __restrict__ dst,
                             const float* __restrict__ hs2, float* __restrict__ agg2) {
    long gid = (long)blockIdx.x * blockDim.x + threadIdx.x;
    int e    = (int)(gid >> 5);
    int lane = (int)(gid & 31);
    if (e < EE) {
        int s = src[e];
        int d = dst[e];
        const float4* hv = (const float4*)(hs2 + (long)s * HH);
        float4 v = hv[lane];                         // coalesced 512B per edge across the wave
        float* ap = agg2 + (long)d * HH + lane * 4;
        atomAddF(ap + 0, v.x);
        atomAddF(ap + 1, v.y);
        atomAddF(ap + 2, v.z);
        atomAddF(ap + 3, v.w);
    }
}

// ---------- per-graph node counts ----------
__global__ void counts_kernel(const int* __restrict__ graph_ids, float* __restrict__ counts) {
    int n = blockIdx.x * blockDim.x + threadIdx.x;
    if (n < NN) atomAddF(&counts[graph_ids[n]], 1.0f);
}

// ---------- layer-2 GEMM via V_WMMA_F32_16X16X4_F32, fused relu + graph pooling ----------
// block = 256 threads (8 waves); block covers 16 node-rows x 128 cols.
// Wave w computes the 16x16 tile at columns [16w, 16w+16). A (16x128, scaled by
// norm_dst) staged in LDS once per block; B = W2 streamed from L2.
__global__ void gemm2_pool_kernel(const float* __restrict__ agg2,
                                  const float* __restrict__ norm_dst,
                                  const float* __restrict__ W2, const float* __restrict__ b2,
                                  const int* __restrict__ graph_ids,
                                  float* __restrict__ hg_sum) {
    __shared__ float Atile[16 * HH];   // 8 KB
    const int tid = threadIdx.x;
    const int m0 = blockIdx.x * 16;

    // cooperative A load: 2048 floats / 256 threads = 8 each, scaled by norm_dst
    for (int i = tid; i < 16 * HH; i += 256) {
        int r = i >> 7;          // row within tile
        int k = i & 127;
        int node = m0 + r;
        float v = 0.0f;
        if (node < NN) v = agg2[(long)node * HH + k] * norm_dst[node];
        Atile[i] = v;
    }
    __syncthreads();

    const int wave = tid >> 5;         // 0..7 -> column tile
    const int lane = tid & 31;
    const int n0   = wave * 16;
    const int mrow = lane & 15;        // A: M index / B,D: N index
    const int kph  = (lane >> 4) << 1; // K phase 0 or 2 (upper-lane half)

    v8f acc = {};
    #pragma unroll 4
    for (int k0 = 0; k0 < HH; k0 += 4) {
        v2f a, b;
        a.x = Atile[mrow * HH + k0 + kph];
        a.y = Atile[mrow * HH + k0 + kph + 1];
        b.x = W2[(long)(k0 + kph) * HH + n0 + mrow];
        b.y = W2[(long)(k0 + kph + 1) * HH + n0 + mrow];
        // 8 args: (neg_a, A, neg_b, B, c_mod, C, reuse_a, reuse_b)
        acc = __builtin_amdgcn_wmma_f32_16x16x4_f32(false, a, false, b,
                                                    (short)0, acc, false, false);
    }

    // epilogue: bias + relu + atomic pool into per-graph accumulator
    const int col  = n0 + mrow;
    const float bias = b2[col];
    const int mhi = (lane >> 4) * 8;   // D layout: VGPR r -> M = r (+8 for upper lanes)
    #pragma unroll
    for (int r = 0; r < 8; ++r) {
        int node = m0 + r + mhi;
        if (node < NN) {
            float v = acc[r] + bias;
            v = v > 0.0f ? v : 0.0f;
            int g = graph_ids[node];
            atomAddF(&hg_sum[g * HH + col], v);
        }
    }
}

// ---------- head: out = (hg_sum / clip(counts,1)) @ Wfc + bfc ----------
__global__ void head_kernel(const float* __restrict__ hg_sum, const float* __restrict__ counts,
                            const float* __restrict__ Wfc, const float* __restrict__ bfc,
                            float* __restrict__ out) {
    int i = blockIdx.x * blockDim.x + threadIdx.x;
    if (i < GG * CC) {
        int g = i / CC;
        int c = i % CC;
        float cnt = counts[g];
        cnt = cnt < 1.0f ? 1.0f : cnt;
        float inv = 1.0f / cnt;
        float s = 0.0f;
        #pragma unroll 8
        for (int k = 0; k < HH; ++k)
            s = fmaf(hg_sum[g * HH + k] * inv, Wfc[k * CC + c], s);
        out[i] = s + bfc[c];
    }
}

extern "C" void kernel_launch(void* const* d_in, const int* in_sizes, int n_in,
                              void* d_out, int out_size, void* d_ws, size_t ws_size,
                              hipStream_t stream) {
    const int*   src       = (const int*)d_in[0];
    const int*   dst       = (const int*)d_in[1];
    const int*   graph_ids = (const int*)d_in[2];
    const float* W1        = (const float*)d_in[3];
    const float* b1        = (const float*)d_in[4];
    const float* W2        = (const float*)d_in[5];
    const float* b2        = (const float*)d_in[6];
    const float* Wfc       = (const float*)d_in[7];
    const float* bfc       = (const float*)d_in[8];
    float* out = (float*)d_out;

    // workspace layout: zeroed region first, then write-only buffers
    float* w        = (float*)d_ws;
    float* in_deg   = w;                        // N
    float* out_deg  = in_deg  + NN;             // N
    float* agg1     = out_deg + NN;             // N
    float* counts   = agg1    + NN;             // G
    float* hg_sum   = counts  + GG;             // G*H
    float* agg2     = hg_sum  + (long)GG * HH;  // N*H   (end of zero region)
    float* norm_src = agg2    + (long)NN * HH;  // N
    float* norm_dst = norm_src + NN;            // N
    float* hs0      = norm_dst + NN;            // N
    float* hs2      = hs0      + NN;            // N*H

    const long n_zero = 3L * NN + GG + (long)GG * HH + (long)NN * HH;

    zero_kernel<<<2048, 256, 0, stream>>>(w, n_zero);

    degree_kernel<<<(EE + 255) / 256, 256, 0, stream>>>(src, dst, in_deg, out_deg);

    norm_kernel<<<(NN + 255) / 256, 256, 0, stream>>>(in_deg, out_deg, norm_src, norm_dst, hs0);

    edge1_kernel<<<(EE + 255) / 256, 256, 0, stream>>>(src, dst, hs0, agg1);

    h1_kernel<<<(NN * HH) / 256, 256, 0, stream>>>(agg1, norm_dst, norm_src, W1, b1, hs2);

    edge2_kernel<<<(long)EE * 32 / 256, 256, 0, stream>>>(src, dst, hs2, agg2);

    counts_kernel<<<(NN + 255) / 256, 256, 0, stream>>>(graph_ids, counts);

    gemm2_pool_kernel<<<(NN + 15) / 16, 256, 0, stream>>>(agg2, norm_dst, W2, b2,
                                                          graph_ids, hg_sum);

    head_kernel<<<(GG * CC + 255) / 256, 256, 0, stream>>>(hg_sum, counts, Wfc, bfc, out);
}